// SimCLRLoss_53961969106974
// MI455X (gfx1250) — compile-verified
//
#include <hip/hip_runtime.h>
#include <math.h>

#define N_PAIRS   4096
#define TWO_N     8192
#define DIM       128
#define BASE_TEMP 0.07f
#define EPS_NRM   1e-12f

#define WPB       4          // waves per block (kernel 2)
#define RPB       (WPB * 16) // rows per block = 64
#define NTHREADS  (WPB * 32) // 128
#define CHUNK     64         // reps rows per LDS buffer (64*128*4 = 32 KB)
#define CHUNK_B   (CHUNK * DIM * 4)

typedef __attribute__((ext_vector_type(2))) float v2f;
typedef __attribute__((ext_vector_type(8))) float v8f;

// ---------------------------------------------------------------------------
// Kernel 1: row-normalize z_i / z_j (wave32: one wave per pair, 4 floats/lane)
// and compute positives pos[k] = dot(zi_hat_k, zj_hat_k).
// ---------------------------------------------------------------------------
__global__ __launch_bounds__(256) void simclr_normalize_kernel(
    const float* __restrict__ zi, const float* __restrict__ zj,
    float* __restrict__ reps, float* __restrict__ pos)
{
    const int pair = (blockIdx.x * blockDim.x + threadIdx.x) >> 5;
    const int lane = threadIdx.x & 31;
    if (pair >= N_PAIRS) return;

    const float4 a = *(const float4*)(zi + pair * DIM + lane * 4);
    const float4 b = *(const float4*)(zj + pair * DIM + lane * 4);

    float ssa = a.x * a.x + a.y * a.y + a.z * a.z + a.w * a.w;
    float ssb = b.x * b.x + b.y * b.y + b.z * b.z + b.w * b.w;
#pragma unroll
    for (int m = 1; m < 32; m <<= 1) {
        ssa += __shfl_xor(ssa, m, 32);
        ssb += __shfl_xor(ssb, m, 32);
    }
    const float ra = 1.0f / fmaxf(sqrtf(ssa), EPS_NRM);
    const float rb = 1.0f / fmaxf(sqrtf(ssb), EPS_NRM);

    float4 an, bn;
    an.x = a.x * ra; an.y = a.y * ra; an.z = a.z * ra; an.w = a.w * ra;
    bn.x = b.x * rb; bn.y = b.y * rb; bn.z = b.z * rb; bn.w = b.w * rb;

    float d = an.x * bn.x + an.y * bn.y + an.z * bn.z + an.w * bn.w;
#pragma unroll
    for (int m = 1; m < 32; m <<= 1) d += __shfl_xor(d, m, 32);

    *(float4*)(reps + pair * DIM + lane * 4)             = an;
    *(float4*)(reps + (pair + N_PAIRS) * DIM + lane * 4) = bn;
    if (lane == 0) pos[pair] = d;
}

// ---------------------------------------------------------------------------
// Async global->LDS staging: 16 x B128 per thread covers one 32 KB chunk
// (128 threads * 16 B * 16 slots). Same immediate offset is applied to both
// the LDS and the global side by the hardware.
// ---------------------------------------------------------------------------
#define ASYNC_CP(l, g, OFF)                                               \
    asm volatile("global_load_async_to_lds_b128 %0, %1, off offset:" #OFF \
                 :: "v"(l), "v"(g) : "memory")

__device__ __forceinline__ void stage_chunk_async(unsigned laddr, const float* gaddr)
{
    ASYNC_CP(laddr, gaddr, 0);
    ASYNC_CP(laddr, gaddr, 2048);
    ASYNC_CP(laddr, gaddr, 4096);
    ASYNC_CP(laddr, gaddr, 6144);
    ASYNC_CP(laddr, gaddr, 8192);
    ASYNC_CP(laddr, gaddr, 10240);
    ASYNC_CP(laddr, gaddr, 12288);
    ASYNC_CP(laddr, gaddr, 14336);
    ASYNC_CP(laddr, gaddr, 16384);
    ASYNC_CP(laddr, gaddr, 18432);
    ASYNC_CP(laddr, gaddr, 20480);
    ASYNC_CP(laddr, gaddr, 22528);
    ASYNC_CP(laddr, gaddr, 24576);
    ASYNC_CP(laddr, gaddr, 26624);
    ASYNC_CP(laddr, gaddr, 28672);
    ASYNC_CP(laddr, gaddr, 30720);
}

// ---------------------------------------------------------------------------
// Kernel 2: fused sim = reps @ reps^T (fp32 WMMA 16x16x4) + masked online
// log-sum-exp per row + loss reduction, with double-buffered async LDS
// staging of the column panel.
//
// Per wave: a 16-row A strip in registers in the WMMA 16x16x4 f32 A layout
// (lanes 0-15 carry K,K+1; lanes 16-31 carry K+2,K+3). B tiles read from LDS
// as v2f per K-step. C layout: VGPR i of lanes 0-15 = row i, lanes 16-31 =
// row i+8, N = lane&15 -> per-lane online (max,sum) LSE state per row.
// ---------------------------------------------------------------------------
__global__ __launch_bounds__(NTHREADS) void simclr_lse_wmma_kernel(
    const float* __restrict__ reps, const float* __restrict__ pos,
    float* __restrict__ out)
{
    __shared__ float lds[2][CHUNK * DIM]; // 2 x 32 KB double buffer

    const int t    = threadIdx.x;
    const int lane = t & 31;
    const int wave = t >> 5;      // 0..3
    const int half = lane >> 4;   // 0: rows i, 1: rows i+8
    const int nl   = lane & 15;   // column within 16-wide tile
    const int r0   = blockIdx.x * RPB + wave * 16;
    const float inv_temp = (float)N_PAIRS / BASE_TEMP;

    // Raw 32-bit LDS offset of the staging buffers for this thread.
    unsigned lds_base;
    {
        __attribute__((address_space(3))) float* p3 =
            (__attribute__((address_space(3))) float*)&lds[0][0];
        lds_base = (unsigned)(unsigned long long)p3;
    }
    const unsigned lthread = lds_base + (unsigned)t * 16u;

    // Load A strip: 16 rows x 128 K, 32 x v2f per lane (64 VGPRs).
    v2f a[32];
    {
        const float* arow = reps + (r0 + nl) * DIM + half * 2;
#pragma unroll
        for (int kk = 0; kk < 32; ++kk)
            a[kk] = *(const v2f*)(arow + kk * 4);
    }

    // Online LSE state for this lane's 8 rows (over its 1/16 column subset).
    float mx[8], sm[8];
#pragma unroll
    for (int i = 0; i < 8; ++i) { mx[i] = -INFINITY; sm[i] = 0.0f; }

    // Prologue: kick off async copy of chunk 0 into buffer 0.
    stage_chunk_async(lthread, reps + t * 4);

    const int n_chunks = TWO_N / CHUNK;
    int p = 0;
    for (int chunk = 0; chunk < n_chunks; ++chunk) {
        // My async copies for chunk -> buf[p] complete, then publish to WG.
        asm volatile("s_wait_asynccnt 0x0" ::: "memory");
        __syncthreads();
        // All waves are past compute(chunk-1) on buf[p^1]: safe to refill it.
        if (chunk + 1 < n_chunks)
            stage_chunk_async(lthread + (unsigned)(p ^ 1) * CHUNK_B,
                              reps + (chunk + 1) * CHUNK * DIM + t * 4);

#pragma unroll
        for (int ct = 0; ct < CHUNK / 16; ++ct) {
            const float* brow = &lds[p][0] + (ct * 16 + nl) * DIM + half * 2;
            v8f c = {};
#pragma unroll
            for (int kk = 0; kk < 32; ++kk) {
                v2f b = *(const v2f*)(brow + kk * 4);
                c = __builtin_amdgcn_wmma_f32_16x16x4_f32(
                        false, a[kk], false, b, (short)0, c, false, false);
            }
            // Online masked LSE update: this lane holds column cg of 8 rows.
            const int cg = chunk * CHUNK + ct * 16 + nl;
#pragma unroll
            for (int i = 0; i < 8; ++i) {
                const int rg = r0 + i + half * 8;
                if (cg == rg) continue;             // mask self-similarity
                const float v  = c[i] * inv_temp;
                const float m2 = fmaxf(mx[i], v);
                sm[i] = sm[i] * __expf(mx[i] - m2) + __expf(v - m2);
                mx[i] = m2;
            }
        }
        p ^= 1;
    }

    // Merge (max,sum) across the 16 lanes sharing each row (butterfly within
    // each 16-lane half; rows differ between halves).
#pragma unroll
    for (int i = 0; i < 8; ++i) {
        float m = mx[i], s = sm[i];
#pragma unroll
        for (int d = 1; d < 16; d <<= 1) {
            const float mo = __shfl_xor(m, d, 32);
            const float so = __shfl_xor(s, d, 32);
            const float m2 = fmaxf(m, mo);
            s = s * __expf(m - m2) + so * __expf(mo - m2);
            m = m2;
        }
        mx[i] = m; sm[i] = s;
    }

    if (nl == 0) {
        float acc = 0.0f;
#pragma unroll
        for (int i = 0; i < 8; ++i) {
            const int rg = r0 + i + half * 8;
            const float lse = mx[i] + __logf(sm[i]);
            const float pv  = pos[rg & (N_PAIRS - 1)]; // pos[rg % N]
            acc += lse - pv * inv_temp;
        }
        atomicAdd(out, acc * (1.0f / (float)TWO_N));
    }
}

// ---------------------------------------------------------------------------
extern "C" void kernel_launch(void* const* d_in, const int* in_sizes, int n_in,
                              void* d_out, int out_size, void* d_ws, size_t ws_size,
                              hipStream_t stream)
{
    (void)in_sizes; (void)n_in; (void)ws_size;
    const float* zi = (const float*)d_in[0];
    const float* zj = (const float*)d_in[1];
    float* out  = (float*)d_out;
    float* reps = (float*)d_ws;                       // 8192*128 f32 = 4 MB
    float* pos  = reps + (size_t)TWO_N * DIM;         // 4096 f32

    hipMemsetAsync(out, 0, (size_t)out_size * sizeof(float), stream);

    simclr_normalize_kernel<<<N_PAIRS / 8, 256, 0, stream>>>(zi, zj, reps, pos);
    simclr_lse_wmma_kernel<<<TWO_N / RPB, NTHREADS, 0, stream>>>(reps, pos, out);
}